// Explore_Recommendation_Decoder_28716151341090
// MI455X (gfx1250) — compile-verified
//
#include <hip/hip_runtime.h>
#include <math.h>

typedef __attribute__((ext_vector_type(2))) float v2f;
typedef __attribute__((ext_vector_type(8))) float v8f;

constexpr int BB   = 256;     // batch (= GEMM M)
constexpr int LL   = 50;      // seq len
constexpr int HH   = 128;     // hidden
constexpr int KDIM = 2 * HH;  // 256 = feat dim (= GEMM K)
constexpr int NTOT = 100000;  // items (= GEMM N)
constexpr int WPR  = (NTOT + 31) / 32;  // 3125 mask words per row
constexpr float NEG_MASK  = -1.0e9f;
constexpr float NEG_LOGIT = -1.0e30f;   // exp(x - max) underflows to exact 0

// ---------------------------------------------------------------------------
// K1: additive attention -> featT (K-major: featT[k][b]) ; build item bitmask.
// One block per batch row, 128 threads (thread h = hidden index).
// ---------------------------------------------------------------------------
__global__ __launch_bounds__(128)
void attn_feat_kernel(const float* __restrict__ all_memory,
                      const float* __restrict__ last_memory,
                      const int* __restrict__ seq_item,
                      const unsigned char* __restrict__ mask,
                      const float* __restrict__ Ue_w, const float* __restrict__ Ue_b,
                      const float* __restrict__ We_w, const float* __restrict__ We_b,
                      const float* __restrict__ Ve_w, const float* __restrict__ Ve_b,
                      float* __restrict__ featT, unsigned int* __restrict__ bits)
{
    const int b = blockIdx.x;
    const int h = threadIdx.x;  // 0..127

    __shared__ float lm_s[HH];
    __shared__ float am_s[10][HH];
    __shared__ float scores_s[LL];
    __shared__ float attn_s[LL];
    __shared__ float red_s[4];

    // zero this row's bitmask, then set bits for items > 0
    unsigned int* rowbits = bits + (size_t)b * WPR;
    for (int w = h; w < WPR; w += 128) rowbits[w] = 0u;
    lm_s[h] = last_memory[b * HH + h];
    __syncthreads();
    if (h < LL) {
        int item = seq_item[b * LL + h];
        if (item > 0) atomicOr(&rowbits[item >> 5], 1u << (item & 31));
    }

    // q_h = (last_memory @ We_w + We_b)[h]
    float q = We_b[h];
    for (int k = 0; k < HH; ++k) q += lm_s[k] * We_w[k * HH + h];

    const float ueb = Ue_b[h];
    const float veh = Ve_w[h];  // Ve_w is [H,1]

    // scores over L in chunks of 10 (register-tiled over l)
    for (int lc = 0; lc < LL; lc += 10) {
        for (int idx = h; idx < 10 * HH; idx += 128) {
            int li = idx >> 7, k = idx & (HH - 1);
            am_s[li][k] = all_memory[((size_t)b * LL + lc + li) * HH + k];
        }
        __syncthreads();
        float acc[10];
#pragma unroll
        for (int li = 0; li < 10; ++li) acc[li] = ueb;
        for (int k = 0; k < HH; ++k) {
            float ue = Ue_w[k * HH + h];
#pragma unroll
            for (int li = 0; li < 10; ++li) acc[li] += am_s[li][k] * ue;
        }
#pragma unroll
        for (int li = 0; li < 10; ++li) {
            float t = tanhf(acc[li] + q) * veh;
            for (int off = 16; off > 0; off >>= 1) t += __shfl_down(t, off, 32);
            if ((threadIdx.x & 31) == 0) red_s[threadIdx.x >> 5] = t;
            __syncthreads();
            if (h == 0)
                scores_s[lc + li] = red_s[0] + red_s[1] + red_s[2] + red_s[3] + Ve_b[0];
            __syncthreads();
        }
    }

    // masked softmax over L (tiny: single thread)
    if (h == 0) {
        float mx = -3.0e38f;
        for (int l = 0; l < LL; ++l) {
            float s = mask[b * LL + l] ? NEG_MASK : scores_s[l];
            scores_s[l] = s;
            mx = fmaxf(mx, s);
        }
        float sum = 0.f;
        for (int l = 0; l < LL; ++l) { float e = expf(scores_s[l] - mx); attn_s[l] = e; sum += e; }
        float inv = 1.f / sum;
        for (int l = 0; l < LL; ++l) attn_s[l] *= inv;
    }
    __syncthreads();

    // ctx_h ; write feat transposed (K-major) for coalesced A-frag loads in K2
    float ctx = 0.f;
    for (int l = 0; l < LL; ++l)
        ctx += attn_s[l] * all_memory[((size_t)b * LL + l) * HH + h];
    featT[(size_t)h * BB + b]        = ctx;      // k = h
    featT[(size_t)(HH + h) * BB + b] = lm_s[h];  // k = H + h
}

// ---------------------------------------------------------------------------
// K2: logits = feat[256,256] @ Wexp[256,100000] via V_WMMA_F32_16X16X4_F32.
// Grid (3125,1), 256 threads = 8 waves. Block tile: 256(M) x 32(N).
// Wexp tile [256K x 32N] staged once in LDS (stride 40 -> conflict-free),
// shared by all 8 waves. Each wave: 32M x 32N via 2x2 WMMA subtiles
// (4 accumulators), A-frags coalesced from L2-resident featT.
// ---------------------------------------------------------------------------
constexpr int BSTR = 40;  // padded N-stride: +2 rows => +80 ≡ +16 banks (disjoint)

__global__ __launch_bounds__(256)
void gemm_mask_kernel(const float* __restrict__ featT,   // [K][M] = [256][256]
                      const float* __restrict__ Wexp,    // [K][N]
                      const unsigned int* __restrict__ bits,
                      float* __restrict__ out)
{
    __shared__ float Bs[KDIM][BSTR];  // 40 KB

    const int n0b = blockIdx.x * 32;

    // stage Wexp[:, n0b .. n0b+31] into LDS (each element read once globally)
    for (int idx = threadIdx.x; idx < KDIM * 32; idx += 256) {
        int k  = idx >> 5;
        int nn = idx & 31;
        Bs[k][nn] = Wexp[(size_t)k * NTOT + n0b + nn];
    }
    __syncthreads();

    const int lane = threadIdx.x & 31;
    const int wave = threadIdx.x >> 5;       // 0..7 -> M strip
    const int koff = (lane >> 4) << 1;       // 0 (lanes 0-15) or 2 (lanes 16-31)
    const int ml   = lane & 15;
    const int m0   = wave * 32;              // wave covers M rows m0..m0+31

    v8f c00 = {0.f,0.f,0.f,0.f,0.f,0.f,0.f,0.f};
    v8f c01 = c00, c10 = c00, c11 = c00;

#pragma unroll 4
    for (int k = 0; k < KDIM; k += 4) {
        const int kk = k + koff;
        v2f a0, a1, b0, b1;
        a0.x = featT[(size_t)kk * BB + m0 + ml];
        a0.y = featT[(size_t)(kk + 1) * BB + m0 + ml];
        a1.x = featT[(size_t)kk * BB + m0 + 16 + ml];
        a1.y = featT[(size_t)(kk + 1) * BB + m0 + 16 + ml];
        b0.x = Bs[kk][ml];      b0.y = Bs[kk + 1][ml];
        b1.x = Bs[kk][16 + ml]; b1.y = Bs[kk + 1][16 + ml];
        c00 = __builtin_amdgcn_wmma_f32_16x16x4_f32(false, a0, false, b0, (short)0, c00, false, false);
        c01 = __builtin_amdgcn_wmma_f32_16x16x4_f32(false, a0, false, b1, (short)0, c01, false, false);
        c10 = __builtin_amdgcn_wmma_f32_16x16x4_f32(false, a1, false, b0, (short)0, c10, false, false);
        c11 = __builtin_amdgcn_wmma_f32_16x16x4_f32(false, a1, false, b1, (short)0, c11, false, false);
    }

    // C/D layout: VGPR r -> M = r (+8 for lanes 16-31), N = lane & 15
    const int mtop = m0 + ((lane >> 4) << 3);
    const v8f* accs[4] = { &c00, &c01, &c10, &c11 };
#pragma unroll
    for (int t = 0; t < 4; ++t) {
        const int msub = (t >> 1) * 16;
        const int nsub = (t & 1) * 16;
        const int n = n0b + nsub + ml;
        const v8f& c = *accs[t];
#pragma unroll
        for (int r = 0; r < 8; ++r) {
            int m = mtop + msub + r;
            unsigned int w = bits[(size_t)m * WPR + (n >> 5)];
            float v = ((w >> (n & 31)) & 1u) ? NEG_LOGIT : c[r];
            out[(size_t)m * NTOT + n] = v;
        }
    }
}

// ---------------------------------------------------------------------------
// K3a: online (max, sum-exp) per row -> rmax, rinv. One block per row.
// ---------------------------------------------------------------------------
__global__ __launch_bounds__(256)
void row_reduce_kernel(const float* __restrict__ out,
                       float* __restrict__ rmax, float* __restrict__ rinv)
{
    const int b = blockIdx.x;
    const int tid = threadIdx.x;
    const float* row = out + (size_t)b * NTOT;

    float m = -3.0e38f, s = 0.f;
    for (int n = tid; n < NTOT; n += 256) {
        float x = row[n];
        if (x > m) { s = s * expf(m - x) + 1.f; m = x; }
        else       { s += expf(x - m); }
    }

    __shared__ float sm[256], ss[256];
    sm[tid] = m; ss[tid] = s;
    __syncthreads();
    for (int off = 128; off > 0; off >>= 1) {
        if (tid < off) {
            float m2 = sm[tid + off], s2 = ss[tid + off];
            float M = fmaxf(sm[tid], m2);
            ss[tid] = ss[tid] * expf(sm[tid] - M) + s2 * expf(m2 - M);
            sm[tid] = M;
        }
        __syncthreads();
    }
    if (tid == 0) { rmax[b] = sm[0]; rinv[b] = 1.f / ss[0]; }
}

// ---------------------------------------------------------------------------
// K3b: normalize in place. Grid = (ceil(N/256), B).
// ---------------------------------------------------------------------------
__global__ __launch_bounds__(256)
void normalize_kernel(float* __restrict__ out,
                      const float* __restrict__ rmax, const float* __restrict__ rinv)
{
    const int b = blockIdx.y;
    const int n = blockIdx.x * 256 + threadIdx.x;
    if (n < NTOT) {
        size_t i = (size_t)b * NTOT + n;
        out[i] = expf(out[i] - rmax[b]) * rinv[b];
    }
}

// ---------------------------------------------------------------------------
extern "C" void kernel_launch(void* const* d_in, const int* in_sizes, int n_in,
                              void* d_out, int out_size, void* d_ws, size_t ws_size,
                              hipStream_t stream)
{
    const float* all_memory  = (const float*)d_in[0];
    const float* last_memory = (const float*)d_in[1];
    const int*   seq_item    = (const int*)d_in[2];
    const unsigned char* mask = (const unsigned char*)d_in[3];
    const float* Ue_w = (const float*)d_in[4];
    const float* Ue_b = (const float*)d_in[5];
    const float* We_w = (const float*)d_in[6];
    const float* We_b = (const float*)d_in[7];
    const float* Ve_w = (const float*)d_in[8];
    const float* Ve_b = (const float*)d_in[9];
    const float* Wexp = (const float*)d_in[10];
    float* out = (float*)d_out;

    // workspace layout
    char* ws = (char*)d_ws;
    float* featT       = (float*)ws;                                    // 256 KB
    unsigned int* bits = (unsigned int*)(ws + (size_t)KDIM * BB * 4);   // 3.2 MB
    float* rmax        = (float*)(ws + (size_t)KDIM * BB * 4 + (size_t)BB * WPR * 4);
    float* rinv        = rmax + BB;

    attn_feat_kernel<<<BB, 128, 0, stream>>>(all_memory, last_memory, seq_item, mask,
                                             Ue_w, Ue_b, We_w, We_b, Ve_w, Ve_b,
                                             featT, bits);

    gemm_mask_kernel<<<dim3(NTOT / 32, 1), 256, 0, stream>>>(featT, Wexp, bits, out);

    row_reduce_kernel<<<BB, 256, 0, stream>>>(out, rmax, rinv);

    normalize_kernel<<<dim3((NTOT + 255) / 256, BB), 256, 0, stream>>>(out, rmax, rinv);
}